// Spline_996432412960
// MI455X (gfx1250) — compile-verified
//
#include <hip/hip_runtime.h>

#define NPTS 10
#define NINT 9

typedef float v4f __attribute__((ext_vector_type(4)));

// ---------------------------------------------------------------------------
// Kernel 1: compute not-a-knot cubic spline coefficients (runs on 1 thread;
// 10x10 solve is negligible). Mirrors the reference construction exactly,
// solved with partial-pivot Gaussian elimination in fp32.
// Output layout in ws: per interval i (0..8): {c3, c2, c1, c0} as float4.
// ---------------------------------------------------------------------------
__global__ void spline_coeff_kernel(const float* __restrict__ points,
                                    const float* __restrict__ values,
                                    float* __restrict__ ws) {
  if (blockIdx.x != 0 || threadIdx.x != 0) return;

  float x[NPTS], y[NPTS];
  for (int i = 0; i < NPTS; ++i) { x[i] = points[i]; y[i] = values[i]; }

  float dx[NINT], sl[NINT];
  for (int i = 0; i < NINT; ++i) {
    dx[i] = x[i + 1] - x[i];
    sl[i] = (y[i + 1] - y[i]) / dx[i];
  }

  // Augmented system A | b  (A: NPTS x NPTS, b in column NPTS)
  float A[NPTS][NPTS + 1];
  for (int i = 0; i < NPTS; ++i)
    for (int j = 0; j <= NPTS; ++j) A[i][j] = 0.0f;

  for (int i = 1; i < NPTS - 1; ++i) {
    A[i][i - 1] = dx[i];
    A[i][i]     = 2.0f * (dx[i - 1] + dx[i]);
    A[i][i + 1] = dx[i - 1];
    A[i][NPTS]  = 3.0f * (dx[i] * sl[i - 1] + dx[i - 1] * sl[i]);
  }
  float d0 = x[2] - x[0];
  A[0][0] = dx[1];
  A[0][1] = d0;
  A[0][NPTS] = ((dx[0] + 2.0f * d0) * dx[1] * sl[0] + dx[0] * dx[0] * sl[1]) / d0;

  float dn = x[NPTS - 1] - x[NPTS - 3];
  A[NPTS - 1][NPTS - 1] = dx[NINT - 2];
  A[NPTS - 1][NPTS - 2] = dn;
  A[NPTS - 1][NPTS] = (dx[NINT - 1] * dx[NINT - 1] * sl[NINT - 2] +
                       (2.0f * dn + dx[NINT - 1]) * dx[NINT - 2] * sl[NINT - 1]) / dn;

  // Gaussian elimination with partial pivoting
  for (int k = 0; k < NPTS; ++k) {
    int piv = k;
    float mx = fabsf(A[k][k]);
    for (int r = k + 1; r < NPTS; ++r) {
      float a = fabsf(A[r][k]);
      if (a > mx) { mx = a; piv = r; }
    }
    if (piv != k) {
      for (int j = k; j <= NPTS; ++j) {
        float tmp = A[k][j]; A[k][j] = A[piv][j]; A[piv][j] = tmp;
      }
    }
    float inv = 1.0f / A[k][k];
    for (int r = k + 1; r < NPTS; ++r) {
      float f = A[r][k] * inv;
      for (int j = k; j <= NPTS; ++j) A[r][j] -= f * A[k][j];
    }
  }
  float s[NPTS];
  for (int k = NPTS - 1; k >= 0; --k) {
    float acc = A[k][NPTS];
    for (int j = k + 1; j < NPTS; ++j) acc -= A[k][j] * s[j];
    s[k] = acc / A[k][k];
  }

  for (int i = 0; i < NINT; ++i) {
    float t  = (s[i] + s[i + 1] - 2.0f * sl[i]) / dx[i];
    float c3 = t / dx[i];
    float c2 = (sl[i] - s[i]) / dx[i] - t;
    ws[4 * i + 0] = c3;
    ws[4 * i + 1] = c2;
    ws[4 * i + 2] = s[i];   // c1
    ws[4 * i + 3] = y[i];   // c0
  }
}

// ---------------------------------------------------------------------------
// Async helpers (CDNA5 gfx1250): HBM -> LDS direct DMA, tracked by ASYNCcnt.
//   LDS address taken per lane from the VDST VGPR; per-lane 16B transfer for B128.
// LDS byte offset = low 32 bits of the flat shared-aperture pointer
// (AMDGPU flat->LDS mapping uses addr[31:0] as the wave-relative offset).
// ---------------------------------------------------------------------------
__device__ __forceinline__ unsigned int lds_off(const void* p) {
  return (unsigned int)(unsigned long long)(uintptr_t)p;
}

__device__ __forceinline__ void async_load_b128(unsigned int lds_byte_off,
                                                const void* gptr) {
  unsigned long long ga = (unsigned long long)(uintptr_t)gptr;
  asm volatile("global_load_async_to_lds_b128 %0, %1, off"
               :
               : "v"(lds_byte_off), "v"(ga)
               : "memory");
}

// ---------------------------------------------------------------------------
// Kernel 2: HBM-roofline streaming evaluation, double-buffered async pipeline.
//  - global_load_async_to_lds_b128 stages next 512B/wave chunk (ASYNCcnt)
//  - s_wait_asynccnt 1 => current buffer complete (async loads retire in order)
//  - ds_load_b128 consume; 8 compares vs SGPR knot thresholds (exact
//    clip(searchsorted(points,x,'right')-1, 0, 8)); per-lane LDS gather of
//    {c3,c2,c1,c0} + knot; Horner cubic; B128 non-temporal store
//  - global_prefetch_b8 two strides ahead warms L2 for the async engine
// Control flow is wave-uniform (iters is grid-uniform; OOB lanes issue
// clamped duplicate loads, only stores are guarded) so ASYNCcnt bookkeeping
// is exact.
// ---------------------------------------------------------------------------
__global__ __launch_bounds__(256) void spline_eval_kernel(
    const float* __restrict__ x, const float* __restrict__ points,
    const float* __restrict__ ws, float* __restrict__ out, int n) {
  __shared__ v4f   stage[2][256];  // async staging, one 16B slot per thread
  __shared__ v4f   sc[NINT];       // per-interval {c3,c2,c1,c0}
  __shared__ float sp[NINT];       // knot positions p[0..8] (idx range 0..8)

  const int t = threadIdx.x;
  if (t < NINT) {
    const v4f* cw = (const v4f*)ws;
    sc[t] = cw[t];
    sp[t] = points[t];
  }
  __syncthreads();

  // Uniform addresses -> scalar (s_load) knot thresholds, held in SGPRs.
  const float p1 = points[1], p2 = points[2], p3 = points[3], p4 = points[4];
  const float p5 = points[5], p6 = points[6], p7 = points[7], p8 = points[8];

  const v4f* __restrict__ x4 = (const v4f*)x;
  v4f* __restrict__ o4 = (v4f*)out;
  const int n4 = n >> 2;
  const int S = gridDim.x * blockDim.x;
  const int gid = blockIdx.x * blockDim.x + t;

  if (n4 > 0) {
    const int iters = (n4 + S - 1) / S;  // grid-uniform trip count
    const unsigned int slot0 = lds_off(&stage[0][t]);
    const unsigned int slot1 = lds_off(&stage[1][t]);

    // Prologue: stage chunk 0 (clamped address for OOB lanes).
    {
      int i0 = gid < n4 ? gid : (n4 - 1);
      async_load_b128(slot0, x4 + i0);
    }

    for (int k = 0; k < iters; ++k) {
      const int i = gid + k * S;

      if (k + 1 < iters) {                     // uniform branch
        int inext = gid + (k + 1) * S;
        if (inext >= n4) inext = n4 - 1;       // clamp, keep EXEC full
        async_load_b128((k & 1) ? slot0 : slot1, x4 + inext);
        int ipf = inext + S;                   // warm L2 two strides ahead
        if (ipf >= n4) ipf = n4 - 1;
        __builtin_prefetch(x4 + ipf, 0, 0);    // global_prefetch_b8
        asm volatile("s_wait_asynccnt 0x1" ::: "memory");  // chunk k ready
      } else {
        asm volatile("s_wait_asynccnt 0x0" ::: "memory");  // last chunk ready
      }

      v4f v = stage[k & 1][t];                 // ds_load_b128
      v4f r;
#pragma unroll
      for (int j = 0; j < 4; ++j) {
        float xv = v[j];
        int idx = (int)(xv >= p1) + (int)(xv >= p2) + (int)(xv >= p3) +
                  (int)(xv >= p4) + (int)(xv >= p5) + (int)(xv >= p6) +
                  (int)(xv >= p7) + (int)(xv >= p8);
        v4f c = sc[idx];                       // ds_load_b128
        float tt = xv - sp[idx];               // ds_load_b32
        r[j] = ((c[0] * tt + c[1]) * tt + c[2]) * tt + c[3];
      }
      if (i < n4)
        __builtin_nontemporal_store(r, &o4[i]);  // global_store_b128 (NT)
    }
  }

  // Tail (n % 4) — not hit for 4096x4096, kept for generality.
  if (blockIdx.x == 0 && t < (n & 3)) {
    int i = (n4 << 2) + t;
    float xv = x[i];
    int idx = (int)(xv >= p1) + (int)(xv >= p2) + (int)(xv >= p3) +
              (int)(xv >= p4) + (int)(xv >= p5) + (int)(xv >= p6) +
              (int)(xv >= p7) + (int)(xv >= p8);
    v4f c = sc[idx];
    float tt = xv - sp[idx];
    out[i] = ((c[0] * tt + c[1]) * tt + c[2]) * tt + c[3];
  }
}

extern "C" void kernel_launch(void* const* d_in, const int* in_sizes, int n_in,
                              void* d_out, int out_size, void* d_ws, size_t ws_size,
                              hipStream_t stream) {
  const float* x      = (const float*)d_in[0];
  const float* points = (const float*)d_in[1];
  const float* values = (const float*)d_in[2];
  float* out = (float*)d_out;
  float* ws  = (float*)d_ws;   // 9 * float4 = 144 bytes of coefficients
  const int n = in_sizes[0];

  // Tiny one-thread solve; same stream guarantees ordering before eval.
  spline_coeff_kernel<<<1, 32, 0, stream>>>(points, values, ws);

  const int threads = 256;                 // 8 wave32s per workgroup
  int n4 = n >> 2;
  if (n4 < 1) n4 = 1;
  int blocks = (n4 + threads - 1) / threads;
  if (blocks > 4096) blocks = 4096;        // grid-stride: ~1M threads, 4 v4f each
  spline_eval_kernel<<<blocks, threads, 0, stream>>>(x, points, ws, out, n);
}